// QuLinear_6794638263035
// MI455X (gfx1250) — compile-verified
//
#include <hip/hip_runtime.h>
#include <hip/hip_bf16.h>

typedef float v2f __attribute__((ext_vector_type(2)));
typedef float v8f __attribute__((ext_vector_type(8)));

#define N_QUBITS 20
#define W_MUL 0.6324555320336759f /* sqrt(2/5) */

struct Masks { unsigned m[N_QUBITS]; };

// ws layout (floats):
//   0    : v[20][4]            (a.re, a.im, b.re, b.im) per qubit
//   128  : A_re[1024]   1152 : A_im[1024]
//   2176 : B_re[1024]   3200 : B_im[1024]
//   8192 : Mre[1<<20] ; +1M : Mim ; +2M : A1re ; +3M : A1im

// ---------------------------------------------------------------- prep ------
__global__ void prep_kernel(const float* __restrict__ x,
                            const float* __restrict__ w,
                            float* __restrict__ ws) {
    __shared__ float U[10][8];  // 2x2 complex, row-major (r,i) pairs
    int t = threadIdx.x;
    if (t < N_QUBITS) {
        float xv = x[t];
        float th = atanf(xv) * 0.5f;          // Ry(arctan x): half angle
        float ph = atanf(xv * xv) * 0.5f;     // Rz(arctan x^2): half angle
        float c = cosf(th), s = sinf(th);
        const float is2 = 0.7071067811865476f;
        float a = (c - s) * is2;              // Ry*H |0> component 0
        float b = (c + s) * is2;              // component 1
        float cp = cosf(ph), sp = sinf(ph);
        ws[t * 4 + 0] = a * cp;   ws[t * 4 + 1] = -a * sp;  // a * e^{-i ph}
        ws[t * 4 + 2] = b * cp;   ws[t * 4 + 3] = b * sp;   // b * e^{+i ph}
    }
    if (t < 10) {
        // U = Rx(w2) * Rz(w1) * Rx(w0), angles scaled by W_MUL, halved per gate
        float a0 = w[3 * t + 0] * (W_MUL * 0.5f);
        float a1 = w[3 * t + 1] * (W_MUL * 0.5f);
        float a2 = w[3 * t + 2] * (W_MUL * 0.5f);
        float c0 = cosf(a0), s0 = sinf(a0);
        float c2 = cosf(a2), s2 = sinf(a2);
        float pr = cosf(a1), pi = -sinf(a1);  // e^{-i a1}
        // M1 = Rz * Rx(a0): row0 *= p, row1 *= conj(p);  Rx = [[c,-is],[-is,c]]
        float m00r = c0 * pr,  m00i = c0 * pi;
        float m01r = s0 * pi,  m01i = -s0 * pr;   // -i s0 * p
        float m10r = -s0 * pi, m10i = -s0 * pr;   // -i s0 * conj(p)
        float m11r = c0 * pr,  m11i = -c0 * pi;   // c0 * conj(p)
        // U = Rx(a2) * M1 ;  (-i s2)*(zr+i zi) = s2*zi - i s2*zr
        U[t][0] = c2 * m00r + s2 * m10i;  U[t][1] = c2 * m00i - s2 * m10r;
        U[t][2] = c2 * m01r + s2 * m11i;  U[t][3] = c2 * m01i - s2 * m11r;
        U[t][4] = s2 * m00i + c2 * m10r;  U[t][5] = -s2 * m00r + c2 * m10i;
        U[t][6] = s2 * m01i + c2 * m11r;  U[t][7] = -s2 * m01r + c2 * m11i;
    }
    __syncthreads();
    // A = U0 x U1 x ... x U4 ; B = U5 x ... x U9   (32x32 complex each)
    for (int e = t; e < 1024; e += blockDim.x) {
        int r = e >> 5, c = e & 31;
        float ar = 1.f, ai = 0.f, br = 1.f, bi = 0.f;
        #pragma unroll
        for (int k = 0; k < 5; ++k) {
            int rb = (r >> (4 - k)) & 1, cb = (c >> (4 - k)) & 1;
            int idx = (rb * 2 + cb) * 2;
            float er = U[k][idx], ei = U[k][idx + 1];
            float nr = ar * er - ai * ei, ni = ar * ei + ai * er;
            ar = nr; ai = ni;
            er = U[5 + k][idx]; ei = U[5 + k][idx + 1];
            nr = br * er - bi * ei; ni = br * ei + bi * er;
            br = nr; bi = ni;
        }
        ws[128 + e] = ar;  ws[1152 + e] = ai;
        ws[2176 + e] = br; ws[3200 + e] = bi;
    }
}

// ------------------------------------------------------------- zero out -----
__global__ void zero_out_kernel(float* __restrict__ o) {
    o[blockIdx.x * blockDim.x + threadIdx.x] = 0.f;
}

// -------------------------------------------- product state after CNOTs -----
__global__ void product_state_kernel(const float* __restrict__ v, Masks mk,
                                     float* __restrict__ mre,
                                     float* __restrict__ mim) {
    unsigned j = blockIdx.x * blockDim.x + threadIdx.x;
    float re = 1.f, im = 0.f;
    #pragma unroll
    for (int q = 0; q < N_QUBITS; ++q) {
        unsigned p = __popc(j & mk.m[q]) & 1u;   // bit q of L(j)
        float cr = p ? v[q * 4 + 2] : v[q * 4 + 0];
        float ci = p ? v[q * 4 + 3] : v[q * 4 + 1];
        float nr = re * cr - im * ci;
        float ni = re * ci + im * cr;
        re = nr; im = ni;
    }
    mre[j] = re;
    mim[j] = im;
}

// ---------------------------------------------------- WMMA frag helpers -----
__device__ __forceinline__ void load_pfrag(const float* __restrict__ Pre,
                                           const float* __restrict__ Pim,
                                           int m_tile, int k0, int lane,
                                           v2f& pr, v2f& pi) {
    int m = m_tile * 16 + (lane & 15);
    int k = k0 + 2 * (lane >> 4);
    pr[0] = Pre[m * 32 + k];     pr[1] = Pre[m * 32 + k + 1];
    pi[0] = Pim[m * 32 + k];     pi[1] = Pim[m * 32 + k + 1];
}

__device__ __forceinline__ void load_qfrag(const float* __restrict__ Q,
                                           int rowstride, int k0, int n0,
                                           int lane, v2f& q) {
    int k = k0 + 2 * (lane >> 4);
    int n = n0 + (lane & 15);
    q[0] = Q[k * rowstride + n];
    q[1] = Q[(k + 1) * rowstride + n];
}

#define WMMA_F32(acc, A, B) \
    acc = __builtin_amdgcn_wmma_f32_16x16x4_f32(false, A, false, B, (short)0, acc, false, false)

// ------------------------------ stage B: amp1[g] = B32 * M[g]  (complex) ----
__global__ void stageB_kernel(const float* __restrict__ Bre,
                              const float* __restrict__ Bim,
                              const float* __restrict__ Mre,
                              const float* __restrict__ Mim,
                              float* __restrict__ A1re,
                              float* __restrict__ A1im) {
    int lane = threadIdx.x & 31;
    int wave = threadIdx.x >> 5;
    int id = blockIdx.x * 8 + wave;        // 4096 tiles
    int m_tile = id & 1;
    int n_tile = (id >> 1) & 63;
    int g = id >> 7;                       // h_hi' slice (0..31)
    const float* Qre = Mre + g * 32 * 1024;
    const float* Qim = Mim + g * 32 * 1024;
    int n0 = n_tile * 16;
    v8f rr = {}, ii = {}, ri = {}, ir = {};
    #pragma unroll
    for (int kk = 0; kk < 8; ++kk) {
        int k0 = kk * 4;
        v2f pr, pim, qr, qi;
        load_pfrag(Bre, Bim, m_tile, k0, lane, pr, pim);
        load_qfrag(Qre, 1024, k0, n0, lane, qr);
        load_qfrag(Qim, 1024, k0, n0, lane, qi);
        WMMA_F32(rr, pr, qr);
        WMMA_F32(ii, pim, qi);
        WMMA_F32(ri, pr, qi);
        WMMA_F32(ir, pim, qr);
    }
    int half = lane >> 4, nl = lane & 15;
    #pragma unroll
    for (int r = 0; r < 8; ++r) {
        float re = rr[r] - ii[r];
        float im = ri[r] + ir[r];
        int m_out = m_tile * 16 + r + 8 * half;
        int col = n0 + nl;
        A1re[(g * 32 + m_out) * 1024 + col] = re;
        A1im[(g * 32 + m_out) * 1024 + col] = im;
    }
}

// ---------- stage A: amp2 = A32 * amp1 ; hidden += |amp2|^2 over l ----------
__global__ void stageA_kernel(const float* __restrict__ Are,
                              const float* __restrict__ Aim,
                              const float* __restrict__ A1re,
                              const float* __restrict__ A1im,
                              float* __restrict__ out) {
    int lane = threadIdx.x & 31;
    int wave = threadIdx.x >> 5;
    int id = blockIdx.x * 8 + wave;        // 4096 tiles
    int m_tile = id & 1;
    int n_tile = id >> 1;                  // 0..2047
    int n0 = n_tile * 16;
    v8f rr = {}, ii = {}, ri = {}, ir = {};
    #pragma unroll
    for (int kk = 0; kk < 8; ++kk) {
        int k0 = kk * 4;
        v2f pr, pim, qr, qi;
        load_pfrag(Are, Aim, m_tile, k0, lane, pr, pim);
        load_qfrag(A1re, 32768, k0, n0, lane, qr);
        load_qfrag(A1im, 32768, k0, n0, lane, qi);
        WMMA_F32(rr, pr, qr);
        WMMA_F32(ii, pim, qi);
        WMMA_F32(ri, pr, qi);
        WMMA_F32(ir, pim, qr);
    }
    int half = lane >> 4, nl = lane & 15;
    int h_lo = n0 >> 10;                   // all 16 cols share h_lo (16 | 1024)
    #pragma unroll
    for (int r = 0; r < 8; ++r) {
        float re = rr[r] - ii[r];
        float im = ri[r] + ir[r];
        float val = re * re + im * im;
        // reduce over the 16 N-lanes of this half (xor masks stay in-half)
        val += __shfl_xor(val, 1);
        val += __shfl_xor(val, 2);
        val += __shfl_xor(val, 4);
        val += __shfl_xor(val, 8);
        if (nl == 0) {
            int h_hi = m_tile * 16 + r + 8 * half;
            atomicAdd(&out[h_hi * 32 + h_lo], val);
        }
    }
}

// ----------------------------------------------------------------- host -----
extern "C" void kernel_launch(void* const* d_in, const int* in_sizes, int n_in,
                              void* d_out, int out_size, void* d_ws, size_t ws_size,
                              hipStream_t stream) {
    const float* x = (const float*)d_in[0];   // (1,20)
    const float* w = (const float*)d_in[1];   // (60,)
    float* out = (float*)d_out;               // (1024,)
    float* ws = (float*)d_ws;

    float* v    = ws;
    float* Are  = ws + 128;
    float* Aim  = ws + 1152;
    float* Bre  = ws + 2176;
    float* Bim  = ws + 3200;
    float* Mre  = ws + 8192;
    float* Mim  = Mre + (1 << 20);
    float* A1re = Mim + (1 << 20);
    float* A1im = A1re + (1 << 20);

    // CNOT layer as GF(2)-linear basis permutation: per-qubit parity masks.
    // L = F_1 o F_2 o ... o F_K  (F_1 = first CNOT applied). Build by
    // iterating the CNOT list in reverse, mask[t] ^= mask[c].
    Masks mk;
    for (int q = 0; q < N_QUBITS; ++q) mk.m[q] = 1u << (19 - q);
    for (int q = N_QUBITS - 1; q >= 0; --q) {
        mk.m[(q + 2) % N_QUBITS] ^= mk.m[q];   // CNOT(q, q+2) (applied 2nd)
        mk.m[(q + 1) % N_QUBITS] ^= mk.m[q];   // CNOT(q, q+1) (applied 1st)
    }

    prep_kernel<<<1, 256, 0, stream>>>(x, w, ws);
    zero_out_kernel<<<4, 256, 0, stream>>>(out);
    product_state_kernel<<<4096, 256, 0, stream>>>(v, mk, Mre, Mim);
    stageB_kernel<<<512, 256, 0, stream>>>(Bre, Bim, Mre, Mim, A1re, A1im);
    stageA_kernel<<<512, 256, 0, stream>>>(Are, Aim, A1re, A1im, out);
}